// BatchNormSPD_15968688407162
// MI455X (gfx1250) — compile-verified
//
#include <hip/hip_runtime.h>
#include <math.h>

// ---------------- configuration ----------------
#define N      64          // matrix dimension
#define NN     (N * N)     // 4096 elements per matrix
#define P      65          // LDS row pitch (floats) -> conflict-free strided lane reads
#define MSZ    (N * P)     // floats per LDS matrix buffer
#define NTH    512         // 16 waves of 32
#define NWAVE  (NTH / 32)
#define NS_ITERS   14      // Newton-Schulz iterations for sqrt/invsqrt
#define LOG_ROOTS  5       // inverse scaling-and-squaring roots for logm
#define LOG_TERMS  8       // Mercator series order
#define EXP_TERMS  10      // Taylor order for expm
#define EXP_SQ     6       // squarings for expm (scale by 1/64)

enum { WS_MNH = 0, WS_MH, WS_WH, WS_WC, WS_WCNH, WS_WCH,
       WS_GACC, WS_GH, WS_GNH, WS_TACC, WS_MEANNH, WS_NMAT };
#define WS_SCAL (WS_NMAT * NN)   // [WS_SCAL]=var accumulator, [WS_SCAL+1]=scale

typedef float v2f __attribute__((ext_vector_type(2)));
typedef float v8f __attribute__((ext_vector_type(8)));

// Convention: every block-cooperative helper ASSUMES the block is synced at
// entry and GUARANTEES it is synced at exit (single s_barrier per helper).
__device__ __forceinline__ void bar() { __syncthreads(); }

// prefetch one 16KB matrix into cache (global_prefetch_b8 path)
__device__ __forceinline__ void prefetch_mat(const float* p) {
  if (threadIdx.x < 128) __builtin_prefetch(p + threadIdx.x * 32, 0, 0);
}

// ---------------- LDS <-> global helpers (dense 64x64 in global, pitch-P in LDS) ----
__device__ __forceinline__ void g2l(float* dst, const float* src) {
  for (int i = threadIdx.x; i < NN; i += NTH) dst[(i >> 6) * P + (i & 63)] = src[i];
  bar();
}
__device__ __forceinline__ void l2g(float* dst, const float* src) {
  for (int i = threadIdx.x; i < NN; i += NTH) dst[i] = src[(i >> 6) * P + (i & 63)];
  bar();  // protect LDS src from being overwritten by subsequent ops
}
__device__ __forceinline__ void l2g_atomic(float* dst, const float* src, float sc) {
  for (int i = threadIdx.x; i < NN; i += NTH)
    atomicAdd(&dst[i], src[(i >> 6) * P + (i & 63)] * sc);
  bar();
}

// ---------------- elementwise matrix helpers (LDS) ----------------
// dst = a*X + c*I   (dst may alias X; element-local, no cross-thread hazard)
__device__ __forceinline__ void ax_plus_cI(float* dst, float a, const float* X, float c) {
  for (int i = threadIdx.x; i < NN; i += NTH) {
    int r = i >> 6, cc = i & 63;
    dst[r * P + cc] = a * X[r * P + cc] + ((r == cc) ? c : 0.0f);
  }
  bar();
}
__device__ __forceinline__ void set_sid(float* dst, float cdiag) {  // dst = cdiag*I
  for (int i = threadIdx.x; i < NN; i += NTH) {
    int r = i >> 6, cc = i & 63;
    dst[r * P + cc] = (r == cc) ? cdiag : 0.0f;
  }
  bar();
}
__device__ __forceinline__ void copy_l(float* dst, const float* src) {
  for (int i = threadIdx.x; i < NN; i += NTH) {
    int idx = (i >> 6) * P + (i & 63);
    dst[idx] = src[idx];
  }
  bar();
}

// ---------------- block reduction: wave32 shuffles + one LDS pass ----------------
__device__ __forceinline__ float block_sum(float v, float* red) {
  for (int o = 16; o > 0; o >>= 1) v += __shfl_down(v, o, 32);
  const int lane = threadIdx.x & 31, wid = threadIdx.x >> 5;
  if (lane == 0) red[wid] = v;
  bar();
  float r;
  if (wid == 0) {
    float s = (lane < NWAVE) ? red[lane] : 0.0f;
    for (int o = NWAVE / 2; o > 0; o >>= 1) s += __shfl_down(s, o, 32);
    if (lane == 0) red[0] = s;
  }
  bar();
  r = red[0];
  bar();
  return r;
}
__device__ __forceinline__ float frob2(const float* X, float* red) {
  float v = 0.0f;
  for (int i = threadIdx.x; i < NN; i += NTH) {
    float x = X[(i >> 6) * P + (i & 63)];
    v += x * x;
  }
  return block_sum(v, red);
}
__device__ __forceinline__ float trace_l(const float* X, float* red) {
  float v = 0.0f;
  if (threadIdx.x < N) v = X[threadIdx.x * P + threadIdx.x];
  return block_sum(v, red);
}

// ---------------- 64x64 fp32 matmul on matrix cores -------------
// C = alpha*(A*B) + cdiag*I, fused into the WMMA C-fragment epilogue.
// 16 waves; wave w owns 16x16 tile (w>>2, w&3); 16 x V_WMMA_F32_16X16X4_F32.
// A-frag (16x4): lanes 0-15 rows, VGPR0/1 = K{0,1}; lanes 16-31 = K{2,3}.
// B-frag (4x16): lanes hold column n, VGPR0/1 = rows K{0,1} (half 1: K{2,3}).
// C/D (16x16):   VGPR r -> row r (+8 for lanes 16-31), col = lane&15.
// All 16 K-step fragments are preloaded into VGPRs (burst of ds loads, one
// wait), then the WMMAs issue back-to-back on two interleaved accumulators
// to break the D->C latency chain.
// NOTE: C must never alias A or B (holds at every call site) -> no entry barrier.
__device__ __forceinline__ void mm64f(float* __restrict__ C,
                                      const float* __restrict__ A,
                                      const float* __restrict__ Bm,
                                      float alpha, float cdiag) {
  const int lane = threadIdx.x & 31;
  const int wid  = threadIdx.x >> 5;
  const int ti   = (wid >> 2) << 4;
  const int tj   = (wid & 3) << 4;
  const int half = lane >> 4;
  const int l15  = lane & 15;
  const float* arow = A + (ti + l15) * P;
  const float* bcol = Bm + tj + l15;

  v2f af[16], bf[16];
#pragma unroll
  for (int k = 0; k < 16; ++k) {
    const int ka = 4 * k + half * 2;
    af[k].x = arow[ka];
    af[k].y = arow[ka + 1];
    bf[k].x = bcol[ka * P];
    bf[k].y = bcol[(ka + 1) * P];
  }
  v8f c0 = {0.f, 0.f, 0.f, 0.f, 0.f, 0.f, 0.f, 0.f};
  v8f c1 = {0.f, 0.f, 0.f, 0.f, 0.f, 0.f, 0.f, 0.f};
#pragma unroll
  for (int k = 0; k < 16; k += 2) {
    c0 = __builtin_amdgcn_wmma_f32_16x16x4_f32(false, af[k], false, bf[k],
                                               (short)0, c0, false, false);
    c1 = __builtin_amdgcn_wmma_f32_16x16x4_f32(false, af[k + 1], false, bf[k + 1],
                                               (short)0, c1, false, false);
  }
  v8f c = c0 + c1;
  const int mb  = ti + half * 8;
  const int col = tj + l15;
  float* crow = C + col;
#pragma unroll
  for (int r = 0; r < 8; ++r)
    crow[(mb + r) * P] = alpha * c[r] + ((mb + r) == col ? cdiag : 0.0f);
  bar();
}
__device__ __forceinline__ void mm64(float* C, const float* A, const float* Bm) {
  mm64f(C, A, Bm, 1.0f, 0.0f);
}

// ---------------- matrix functions (all matmul-based) ----------------
// In: b[0] = SPD A (destroyed). Out: b[0] = A^{1/2}, b[1] = A^{-1/2}.
__device__ void ns_sqrt_fixed(float* const* b, float* red) {
  float* Y = b[0];
  float* Z = b[1];
  float* W = b[2];
  float* U = b[3];
  float nf = sqrtf(frob2(Y, red));          // Frobenius normalization
  ax_plus_cI(Y, 1.0f / nf, Y, 0.0f);
  set_sid(Z, 1.0f);
#pragma unroll 1
  for (int it = 0; it < NS_ITERS; ++it) {
    mm64f(W, Z, Y, -0.5f, 1.5f);             // W = 1.5I - 0.5*Z*Y   (fused)
    mm64(U, Y, W);                           // Y' = Y*W
    { float* t = Y; Y = U; U = t; }
    mm64(U, W, Z);                           // Z' = W*Z
    { float* t = Z; Z = U; U = t; }
  }
  float sr = sqrtf(nf);
  ax_plus_cI(Y, sr, Y, 0.0f);                // sqrt(A)
  ax_plus_cI(Z, 1.0f / sr, Z, 0.0f);         // invsqrt(A)
  copy_l(b[4], Y);                           // canonicalize: b0=sqrt, b1=invsqrt
  if (Z != b[1]) copy_l(b[1], Z);
  copy_l(b[0], b[4]);
}

// In: b[0] = SPD A (destroyed). Out: b[0] = log(A).
__device__ void logm_fixed(float* const* b, float* red) {
  float tr = trace_l(b[0], red) * (1.0f / (float)N);
  ax_plus_cI(b[0], 1.0f / tr, b[0], 0.0f);   // eigenvalues centered near 1
#pragma unroll 1
  for (int j = 0; j < LOG_ROOTS; ++j) ns_sqrt_fixed(b, red);  // b0 = b0^(1/32)
  ax_plus_cI(b[0], 1.0f, b[0], -1.0f);       // E = R - I
  float* E = b[0];
  float* H = b[1];
  float* D = b[2];
  const float coef[LOG_TERMS] = {1.0f, -0.5f, 1.0f / 3.0f, -0.25f,
                                 0.2f, -1.0f / 6.0f, 1.0f / 7.0f, -0.125f};
  set_sid(H, coef[LOG_TERMS - 1]);
#pragma unroll 1
  for (int i = LOG_TERMS - 2; i >= 0; --i) {
    mm64f(D, E, H, 1.0f, coef[i]);           // D = E*H + coef_i*I   (fused)
    float* t = H; H = D; D = t;
  }
  mm64(D, E, H);                             // log(I+E)
  ax_plus_cI(D, (float)(1 << LOG_ROOTS), D, logf(tr));  // 2^k * . + ln(tr) I
  copy_l(b[0], D);
}

// In: b[0] = symmetric A (destroyed). Out: b[0] = exp(A).
__device__ void expm_fixed(float* const* b, float* red) {
  (void)red;
  ax_plus_cI(b[0], 1.0f / (float)(1 << EXP_SQ), b[0], 0.0f);
  float* A = b[0];
  float* H = b[1];
  float* D = b[2];
  set_sid(H, 1.0f);
#pragma unroll 1
  for (int i = EXP_TERMS; i >= 1; --i) {
    mm64f(D, A, H, 1.0f / (float)i, 1.0f);   // H' = I + (A*H)/i   (fused)
    float* t = H; H = D; D = t;
  }
#pragma unroll 1
  for (int j = 0; j < EXP_SQ; ++j) {
    mm64(D, H, H);
    float* t = H; H = D; D = t;
  }
  copy_l(b[0], H);
}

// b[0] = b[3] * b[0] * b[3]   (congruence with symmetric b[3]; uses b[1])
__device__ __forceinline__ void cong_b3(float* const* b) {
  mm64(b[1], b[3], b[0]);
  mm64(b[0], b[1], b[3]);
}

#define SMEM_DECL                                                   \
  extern __shared__ float smem[];                                   \
  float* b0 = smem;                                                 \
  float* b1 = smem + MSZ;                                           \
  float* b2 = smem + 2 * MSZ;                                       \
  float* b3 = smem + 3 * MSZ;                                       \
  float* b4 = smem + 4 * MSZ;                                       \
  float* red = smem + 5 * MSZ;                                      \
  float* bufs[5] = {b0, b1, b2, b3, b4};                            \
  (void)b2; (void)b4;

// ---------------- kernels ----------------
__global__ void k_zero(float* p, int cnt) {
  int i = blockIdx.x * blockDim.x + threadIdx.x;
  if (i < cnt) p[i] = 0.0f;
}

// Mh/Mnh, Wh (=W^0.5=Wp), Wc = Mnh Wh Mnh, Wch/Wcnh
__global__ void k_params(const float* weight, const float* M, float* ws) {
  SMEM_DECL;
  g2l(b0, M);
  ns_sqrt_fixed(bufs, red);
  l2g(ws + WS_MH * NN, b0);
  l2g(ws + WS_MNH * NN, b1);
  g2l(b0, weight);
  ns_sqrt_fixed(bufs, red);
  l2g(ws + WS_WH * NN, b0);
  g2l(b3, ws + WS_MNH * NN);
  cong_b3(bufs);                              // Wc = Mnh * Wh * Mnh
  l2g(ws + WS_WC * NN, b0);
  ns_sqrt_fixed(bufs, red);
  l2g(ws + WS_WCH * NN, b0);
  l2g(ws + WS_WCNH * NN, b1);
}

// Xc_i = Mnh * sqrt(X_i) * Mnh -> out; accumulate arithmetic mean.
__global__ void k_xc(const float* X, float* out, float* ws, float invB) {
  SMEM_DECL;
  prefetch_mat(ws + WS_MNH * NN);             // warm cache for post-sqrt congruence
  const size_t off = (size_t)blockIdx.x * NN;
  g2l(b0, X + off);
  ns_sqrt_fixed(bufs, red);                   // Xp = X^0.5
  g2l(b3, ws + WS_MNH * NN);
  cong_b3(bufs);                              // Xc
  l2g(out + off, b0);
  l2g_atomic(ws + WS_GACC * NN, b0, invB);
}

// G = mean(Xc); Gh/Gnh
__global__ void k_g(float* ws) {
  SMEM_DECL;
  g2l(b0, ws + WS_GACC * NN);
  ns_sqrt_fixed(bufs, red);
  l2g(ws + WS_GH * NN, b0);
  l2g(ws + WS_GNH * NN, b1);
}

// accumulate mean of log(Gnh Xc Gnh)
__global__ void k_tan(const float* out, float* ws, float invB) {
  SMEM_DECL;
  prefetch_mat(ws + WS_GNH * NN);
  g2l(b0, out + (size_t)blockIdx.x * NN);
  g2l(b3, ws + WS_GNH * NN);
  cong_b3(bufs);
  logm_fixed(bufs, red);
  l2g_atomic(ws + WS_TACC * NN, b0, invB);
}

// mean = Gh exp(Tbar) Gh; store mean^{-1/2}
__global__ void k_mean(float* ws) {
  SMEM_DECL;
  g2l(b0, ws + WS_TACC * NN);
  expm_fixed(bufs, red);
  g2l(b3, ws + WS_GH * NN);
  cong_b3(bufs);                              // Karcher mean
  ns_sqrt_fixed(bufs, red);
  l2g(ws + WS_MEANNH * NN, b1);               // mean^{-1/2}
}

// T_i = log(mnh Xc_i mnh) -> out (in place); accumulate variance
__global__ void k_logT(float* out, float* ws, float invB) {
  SMEM_DECL;
  prefetch_mat(ws + WS_MEANNH * NN);
  const size_t off = (size_t)blockIdx.x * NN;
  g2l(b0, out + off);
  g2l(b3, ws + WS_MEANNH * NN);
  cong_b3(bufs);
  logm_fixed(bufs, red);
  l2g(out + off, b0);
  float f2 = frob2(b0, red);
  if (threadIdx.x == 0) atomicAdd(ws + WS_SCAL, f2 * invB);
}

__global__ void k_scale(const float* shift, float* ws) {
  if (threadIdx.x == 0 && blockIdx.x == 0)
    ws[WS_SCAL + 1] = shift[0] / sqrtf(ws[WS_SCAL] + 1e-5f);
}

// Y = (Mh * Wch exp(Wcnh (Wh (s*T) Wh) Wcnh) Wch * Mh)^2
__global__ void k_out(float* out, const float* ws) {
  SMEM_DECL;
  prefetch_mat(ws + WS_WH * NN);
  prefetch_mat(ws + WS_WCNH * NN);
  const float sc = ws[WS_SCAL + 1];
  float* yi = out + (size_t)blockIdx.x * NN;
  g2l(b0, yi);
  ax_plus_cI(b0, sc, b0, 0.0f);               // scale tangent
  g2l(b3, ws + WS_WH * NN);   cong_b3(bufs);  // S = Wh T Wh
  g2l(b3, ws + WS_WCNH * NN); cong_b3(bufs);  // Wcnh S Wcnh
  expm_fixed(bufs, red);
  prefetch_mat(ws + WS_WCH * NN);
  prefetch_mat(ws + WS_MH * NN);
  g2l(b3, ws + WS_WCH * NN);  cong_b3(bufs);  // Wch . Wch
  g2l(b3, ws + WS_MH * NN);   cong_b3(bufs);  // Mh . Mh
  mm64(b1, b0, b0);                           // sym_pow(., 2)
  l2g(yi, b1);
}

// ---------------- host entry ----------------
extern "C" void kernel_launch(void* const* d_in, const int* in_sizes, int n_in,
                              void* d_out, int out_size, void* d_ws, size_t ws_size,
                              hipStream_t stream) {
  const float* X      = (const float*)d_in[0];
  const float* weight = (const float*)d_in[1];
  const float* M      = (const float*)d_in[2];
  const float* shift  = (const float*)d_in[3];
  float* out = (float*)d_out;
  float* ws  = (float*)d_ws;

  const int Bn = in_sizes[0] / NN;            // 4096 matrices
  const float invB = 1.0f / (float)Bn;
  const size_t sh = (size_t)(5 * MSZ + NTH) * sizeof(float);  // ~85 KB dyn LDS

  // allow >64KB dynamic LDS on the WGP (320 KB available)
  hipFuncSetAttribute((const void*)k_params, hipFuncAttributeMaxDynamicSharedMemorySize, (int)sh);
  hipFuncSetAttribute((const void*)k_xc,     hipFuncAttributeMaxDynamicSharedMemorySize, (int)sh);
  hipFuncSetAttribute((const void*)k_g,      hipFuncAttributeMaxDynamicSharedMemorySize, (int)sh);
  hipFuncSetAttribute((const void*)k_tan,    hipFuncAttributeMaxDynamicSharedMemorySize, (int)sh);
  hipFuncSetAttribute((const void*)k_mean,   hipFuncAttributeMaxDynamicSharedMemorySize, (int)sh);
  hipFuncSetAttribute((const void*)k_logT,   hipFuncAttributeMaxDynamicSharedMemorySize, (int)sh);
  hipFuncSetAttribute((const void*)k_out,    hipFuncAttributeMaxDynamicSharedMemorySize, (int)sh);

  const int zc = WS_NMAT * NN + 2;            // small matrices + scalars
  k_zero<<<(zc + NTH - 1) / NTH, NTH, 0, stream>>>(ws, zc);
  k_params<<<1, NTH, sh, stream>>>(weight, M, ws);
  k_xc    <<<Bn, NTH, sh, stream>>>(X, out, ws, invB);
  k_g     <<<1, NTH, sh, stream>>>(ws);
  k_tan   <<<Bn, NTH, sh, stream>>>(out, ws, invB);
  k_mean  <<<1, NTH, sh, stream>>>(ws);
  k_logT  <<<Bn, NTH, sh, stream>>>(out, ws, invB);
  k_scale <<<1, 32, 0, stream>>>(shift, ws);
  k_out   <<<Bn, NTH, sh, stream>>>(out, ws);
}